// MSDeformAttn_33921651704057
// MI455X (gfx1250) — compile-verified
//
#include <hip/hip_runtime.h>

// Problem constants (match reference)
#define D_MODEL  256
#define N_HEADS  8
#define N_LEVELS 4
#define N_POINTS 4
#define DH       32          // D_MODEL / N_HEADS
#define BATCH    4
#define LEN_IN   21760
#define LEN_Q    21760

typedef __attribute__((ext_vector_type(2))) float v2f;
typedef __attribute__((ext_vector_type(8))) float v8f;

// -----------------------------------------------------------------------------
// Fully-fp32 WMMA GEMM:  C[M,N] = A[M,K] @ W[K,N] + bias[N],  K == 256.
// grid.x = M/128 (8 waves x 16 rows), grid.y = N/64.
// The K x 64 weight tile is cooperatively staged in LDS once per block in a
// pair-packed layout [kpair][n][2] so each B fragment is one ds_load_b64.
// Bias is folded into the accumulator init. A row is prefetched while the
// block fills LDS (global_prefetch_b8 overlaps the staging barrier).
// -----------------------------------------------------------------------------
__global__ __launch_bounds__(256) void gemm_bias_wmma_f32(
    const float* __restrict__ A, const float* __restrict__ W,
    const float* __restrict__ bias, float* __restrict__ C,
    int K, int N) {
  // K fixed at 256: 128 k-pairs x 64 cols x 2 floats = 64 KB
  __shared__ float bsm[128 * 64 * 2];

  const int tid  = threadIdx.x;
  const int wave = tid >> 5;
  const int lane = tid & 31;
  const int lm   = lane & 15;          // col (B/C) or row (A) within 16
  const int lh   = lane >> 4;          // half-wave select
  const int row0 = blockIdx.x * 128 + wave * 16;
  const int col0 = blockIdx.y * 64;
  const int m    = row0 + lm;

  const float* __restrict__ arow = A + (size_t)m * K;

  // Prefetch this lane's A row (K*4 = 1KB = 8 cachelines) while B is staged.
  #pragma unroll
  for (int pf = 0; pf < 8; ++pf)
    __builtin_prefetch(arow + pf * 32, 0, 3);

  // ---- cooperative B-tile stage: tasks = kpair (128) x col-group-of-4 (16)
  for (int t = tid; t < 128 * 16; t += 256) {
    const int n4 = (t & 15) * 4;
    const int kp = t >> 4;
    const float* __restrict__ w0 = W + (size_t)(2 * kp) * N + col0 + n4;
    const float4 r0 = *(const float4*)(w0);
    const float4 r1 = *(const float4*)(w0 + N);
    float* __restrict__ dst = &bsm[(kp * 64 + n4) * 2];
    *(float4*)(dst)     = make_float4(r0.x, r1.x, r0.y, r1.y);
    *(float4*)(dst + 4) = make_float4(r0.z, r1.z, r0.w, r1.w);
  }
  __syncthreads();

  // ---- accumulators initialized with bias (result = bias + A@W)
  const float bb0 = bias[col0 + lm +  0];
  const float bb1 = bias[col0 + lm + 16];
  const float bb2 = bias[col0 + lm + 32];
  const float bb3 = bias[col0 + lm + 48];
  v8f c0, c1, c2, c3;
  #pragma unroll
  for (int r = 0; r < 8; ++r) { c0[r] = bb0; c1[r] = bb1; c2[r] = bb2; c3[r] = bb3; }

  for (int k = 0; k < K; k += 4) {
    // A 16x4 fragment: lane<16 -> K = k,k+1 ; lane>=16 -> K = k+2,k+3
    const int kb = k + (lh << 1);
    const float2 av = *(const float2*)(arow + kb);
    v2f a; a.x = av.x; a.y = av.y;
    // B fragments from LDS: kpair = k/2 + lh, contiguous 8B per fragment,
    // column groups at +0/+128/+256/+384 bytes (immediate ds offsets)
    const float* __restrict__ bp = &bsm[((k >> 1) + lh) * 128 + lm * 2];
    const v2f b0 = *(const v2f*)(bp +  0);
    const v2f b1 = *(const v2f*)(bp + 32);
    const v2f b2 = *(const v2f*)(bp + 64);
    const v2f b3 = *(const v2f*)(bp + 96);
    c0 = __builtin_amdgcn_wmma_f32_16x16x4_f32(false, a, false, b0, (short)0, c0, false, false);
    c1 = __builtin_amdgcn_wmma_f32_16x16x4_f32(false, a, false, b1, (short)0, c1, false, false);
    c2 = __builtin_amdgcn_wmma_f32_16x16x4_f32(false, a, false, b2, (short)0, c2, false, false);
    c3 = __builtin_amdgcn_wmma_f32_16x16x4_f32(false, a, false, b3, (short)0, c3, false, false);
  }

  // C/D layout: vgpr r -> row r (lanes 0-15) / row r+8 (lanes 16-31), col = lm
  #pragma unroll
  for (int r = 0; r < 8; ++r) {
    const int mr = row0 + r + (lh << 3);
    float* __restrict__ crow = C + (size_t)mr * N + col0 + lm;
    crow[0]  = c0[r];
    crow[16] = c1[r];
    crow[32] = c2[r];
    crow[48] = c3[r];
  }
}

// -----------------------------------------------------------------------------
// Fused softmax + sampling-location + bilinear-gather core.
// One wave32 per (b, q, head): lane = channel d in Dh=32, so every corner
// gather is one coalesced 128B load and all control flow is wave-uniform.
// -----------------------------------------------------------------------------
__global__ __launch_bounds__(256) void msda_core_kernel(
    const float* __restrict__ value,     // [B, LEN_IN, 8, 32]
    const float* __restrict__ refpts,    // [B, LEN_Q, 4, 2]
    const float* __restrict__ off_raw,   // [B, LEN_Q, 256] = [.., M, L, P, 2]
    const float* __restrict__ attn_raw,  // [B, LEN_Q, 128] = [.., M, L, P]
    const int*   __restrict__ spatial,   // [4, 2]  (H, W)
    const int*   __restrict__ lvl_start, // [4]
    float* __restrict__ out_core)        // [B, LEN_Q, 256]
{
  const int  wave = threadIdx.x >> 5;
  const int  lane = threadIdx.x & 31;
  const long gw   = (long)blockIdx.x * 8 + wave;   // = (b*LEN_Q + q)*8 + h
  const int  h    = (int)(gw & 7);
  const long bq   = gw >> 3;                       // b*LEN_Q + q
  const int  b    = (int)(bq / LEN_Q);

  // ---- softmax over 16 logits for this (b,q,h): butterfly in half-waves
  const int   j16 = lane & 15;
  float t = attn_raw[bq * 128 + h * 16 + j16];
  float mx = t;
  #pragma unroll
  for (int msk = 8; msk >= 1; msk >>= 1) mx = fmaxf(mx, __shfl_xor(mx, msk, 32));
  float e = expf(t - mx);
  float s = e;
  #pragma unroll
  for (int msk = 8; msk >= 1; msk >>= 1) s += __shfl_xor(s, msk, 32);
  const float wgt = e / s;                         // lane j holds weight(j)

  // ---- per-lane sampling offsets: 32 values = 16 (x,y) pairs
  const float ov = off_raw[bq * 256 + h * 32 + lane];

  float acc = 0.0f;
  // value[b, idx, h, lane] ; stride per idx = 256 floats
  const float* __restrict__ vbase =
      value + (size_t)b * LEN_IN * (N_HEADS * DH) + h * DH + lane;

  #pragma unroll
  for (int l = 0; l < N_LEVELS; ++l) {
    const int   H  = spatial[2 * l + 0];
    const int   W  = spatial[2 * l + 1];
    const int   st = lvl_start[l];
    const float Hf = (float)H, Wf = (float)W;
    const float rx = refpts[(bq * N_LEVELS + l) * 2 + 0];
    const float ry = refpts[(bq * N_LEVELS + l) * 2 + 1];
    #pragma unroll
    for (int p = 0; p < N_POINTS; ++p) {
      const int   j  = l * N_POINTS + p;
      const float ox = __shfl(ov,  2 * j,     32);
      const float oy = __shfl(ov,  2 * j + 1, 32);
      const float w  = __shfl(wgt, j,         32);
      // grid_sample, align_corners=False: pixel = loc*size - 0.5
      const float x = (rx + ox / Wf) * Wf - 0.5f;
      const float y = (ry + oy / Hf) * Hf - 0.5f;
      const float x0f = floorf(x), y0f = floorf(y);
      const float fx = x - x0f, fy = y - y0f;
      const int   x0 = (int)x0f, y0 = (int)y0f;
      const int   x1 = x0 + 1,   y1 = y0 + 1;
      const float w00 = (1.0f - fx) * (1.0f - fy);
      const float w01 = fx * (1.0f - fy);
      const float w10 = (1.0f - fx) * fy;
      const float w11 = fx * fy;
      // all comparisons wave-uniform -> no divergence; zeros padding OOB
      if ((unsigned)x0 < (unsigned)W && (unsigned)y0 < (unsigned)H)
        acc += w * w00 * vbase[(size_t)(st + y0 * W + x0) * 256];
      if ((unsigned)x1 < (unsigned)W && (unsigned)y0 < (unsigned)H)
        acc += w * w01 * vbase[(size_t)(st + y0 * W + x1) * 256];
      if ((unsigned)x0 < (unsigned)W && (unsigned)y1 < (unsigned)H)
        acc += w * w10 * vbase[(size_t)(st + y1 * W + x0) * 256];
      if ((unsigned)x1 < (unsigned)W && (unsigned)y1 < (unsigned)H)
        acc += w * w11 * vbase[(size_t)(st + y1 * W + x1) * 256];
    }
  }
  out_core[bq * 256 + h * 32 + lane] = acc;
}

// -----------------------------------------------------------------------------
extern "C" void kernel_launch(void* const* d_in, const int* in_sizes, int n_in,
                              void* d_out, int out_size, void* d_ws, size_t ws_size,
                              hipStream_t stream) {
  (void)in_sizes; (void)n_in; (void)out_size; (void)ws_size;

  const float* query   = (const float*)d_in[0];   // [4,21760,256]
  const float* refpts  = (const float*)d_in[1];   // [4,21760,4,2]
  const float* inp     = (const float*)d_in[2];   // [4,21760,256]
  const int*   spatial = (const int*)  d_in[3];   // [4,2]
  const int*   lvlst   = (const int*)  d_in[4];   // [4]
  const float* W_off   = (const float*)d_in[5];   // [256,256]
  const float* b_off   = (const float*)d_in[6];   // [256]
  const float* W_attn  = (const float*)d_in[7];   // [256,128]
  const float* b_attn  = (const float*)d_in[8];   // [128]
  const float* W_val   = (const float*)d_in[9];   // [256,256]
  const float* b_val   = (const float*)d_in[10];  // [256]
  const float* W_outp  = (const float*)d_in[11];  // [256,256]
  const float* b_outp  = (const float*)d_in[12];  // [256]
  float*       out     = (float*)d_out;           // [4,21760,256]

  const int Mtot = BATCH * LEN_Q;                 // 87040 (divisible by 128)

  // workspace layout (floats)
  float* value    = (float*)d_ws;                       // Mtot*256
  float* off_raw  = value    + (size_t)Mtot * 256;      // Mtot*256
  float* attn_raw = off_raw  + (size_t)Mtot * 256;      // Mtot*128
  float* core     = attn_raw + (size_t)Mtot * 128;      // Mtot*256

  const dim3 blk(256);

  // 1) value = input_flatten @ W_val + b_val
  gemm_bias_wmma_f32<<<dim3(Mtot / 128, D_MODEL / 64), blk, 0, stream>>>(
      inp, W_val, b_val, value, D_MODEL, D_MODEL);

  // 2) off_raw = query @ W_off + b_off
  gemm_bias_wmma_f32<<<dim3(Mtot / 128, D_MODEL / 64), blk, 0, stream>>>(
      query, W_off, b_off, off_raw, D_MODEL, D_MODEL);

  // 3) attn_raw = query @ W_attn + b_attn   (N = 128)
  gemm_bias_wmma_f32<<<dim3(Mtot / 128, 128 / 64), blk, 0, stream>>>(
      query, W_attn, b_attn, attn_raw, D_MODEL, 128);

  // 4) fused softmax + deformable bilinear gather core
  msda_core_kernel<<<dim3(Mtot), blk, 0, stream>>>(
      value, refpts, off_raw, attn_raw, spatial, lvlst, core);

  // 5) out = core @ W_out + b_out
  gemm_bias_wmma_f32<<<dim3(Mtot / 128, D_MODEL / 64), blk, 0, stream>>>(
      core, W_outp, b_outp, out, D_MODEL, D_MODEL);
}